// SmoothAP_68745246540426
// MI455X (gfx1250) — compile-verified
//
#include <hip/hip_runtime.h>
#include <hip/hip_bf16.h>

// SmoothAP on MI455X (gfx1250, wave32).
//
// Key algorithmic reduction: the reference only consumes the diagonal
// NUM_ID x G x G blocks of sim_all_rk, so the B^3 sigmoid tensor collapses
// to NUM_ID*G*G * B = 1.18M sigmoid evaluations. sim_pos is exactly the
// diagonal 8x8 blocks of sim_all, so a single 384x384x512 FP32 GEMM feeds
// everything. The GEMM runs on the CDNA5 matrix path (V_WMMA_F32_16X16X4_F32).

#define B_SZ   384
#define FEAT   512
#define NUM_ID 48
#define G_SZ   8
#define N_TRIPLE (NUM_ID * G_SZ * G_SZ)   // 3072

typedef __attribute__((ext_vector_type(2))) float v2f;
typedef __attribute__((ext_vector_type(8))) float v8f;

// ---------------------------------------------------------------------------
// Kernel 1: sim_all = preds @ preds^T  (384x384, K=512, fp32)
// One wave (32 threads) per 16x16 output tile; grid = 24x24 tiles.
// A fragment (16x4 fp32): lanes 0-15 hold K={k,k+1} in v[0],v[1];
//                         lanes 16-31 hold K={k+2,k+3}.
// B fragment (4x16) mirrors it with the column row of preds (B = preds^T).
// C/D (16x16 fp32, 8 VGPRs): vgpr r -> M=r (lanes 0-15) / M=r+8 (lanes 16-31).
// ---------------------------------------------------------------------------
__global__ void __launch_bounds__(32)
simall_wmma_kernel(const float* __restrict__ preds, float* __restrict__ sim)
{
    const int ti   = blockIdx.x;        // row tile  0..23
    const int tj   = blockIdx.y;        // col tile  0..23
    const int lane = threadIdx.x;       // 0..31 (wave32)
    const int half = lane >> 4;         // 0: lanes 0-15, 1: lanes 16-31
    const int l16  = lane & 15;

    const float* __restrict__ pa = preds + (ti * 16 + l16) * FEAT;
    const float* __restrict__ pb = preds + (tj * 16 + l16) * FEAT;

    v8f c = {};
    #pragma unroll 8
    for (int k = 0; k < FEAT; k += 4) {
        const int kb = k + half * 2;                 // even -> 8B aligned
        v2f a = *(const v2f*)(pa + kb);              // global_load_b64
        v2f b = *(const v2f*)(pb + kb);
        // 8 args: (neg_a, A, neg_b, B, c_mod, C, reuse_a, reuse_b)
        c = __builtin_amdgcn_wmma_f32_16x16x4_f32(
                false, a, false, b, (short)0, c, false, false);
    }

    const int col   = tj * 16 + l16;
    const int rbase = ti * 16 + half * 8;
    #pragma unroll
    for (int r = 0; r < 8; ++r)
        sim[(rbase + r) * B_SZ + col] = c[r];
}

// ---------------------------------------------------------------------------
// Temperature sigmoid: 1 / (1 + exp(clip(-t/0.01, -50, 50)))
// ---------------------------------------------------------------------------
__device__ __forceinline__ float tsig(float t)
{
    float e = -t * 100.0f;
    e = fminf(fmaxf(e, -50.0f), 50.0f);
    return 1.0f / (1.0f + __expf(e));
}

// ---------------------------------------------------------------------------
// Kernel 2: one wave per (n, i, j) triple (3072 waves total).
//   block_rk = 1 + sum_{k != jg} tsig(sim[ig,k] - sim[ig,jg])   (k over 384)
//   pos_rk   = 1 + sum_{k != j } tsig(sim[ig,n*8+k] - sim[ig,jg]) (k over 8)
//   partial[wid] = pos_rk / block_rk
// ---------------------------------------------------------------------------
__global__ void __launch_bounds__(256)
rank_kernel(const float* __restrict__ sim, float* __restrict__ partial)
{
    const int wid  = (blockIdx.x * blockDim.x + threadIdx.x) >> 5; // 0..3071
    const int lane = threadIdx.x & 31;

    const int j  = wid & (G_SZ - 1);
    const int i  = (wid >> 3) & (G_SZ - 1);
    const int n  = wid >> 6;
    const int ig = n * G_SZ + i;
    const int jg = n * G_SZ + j;

    const float* __restrict__ row = sim + ig * B_SZ;
    const float sij = row[jg];

    // 384 k-values, 12 per lane
    float acc = 0.0f;
    #pragma unroll
    for (int kk = 0; kk < B_SZ / 32; ++kk) {
        const int k = lane + kk * 32;
        const float v = tsig(row[k] - sij);
        acc += (k == jg) ? 0.0f : v;
    }
    // wave32 butterfly reduction
    #pragma unroll
    for (int off = 16; off > 0; off >>= 1)
        acc += __shfl_xor(acc, off, 32);

    if (lane == 0) {
        const float block_rk = acc + 1.0f;
        float pos = 0.0f;
        #pragma unroll
        for (int k = 0; k < G_SZ; ++k)
            if (k != j) pos += tsig(row[n * G_SZ + k] - sij);
        const float pos_rk = pos + 1.0f;
        partial[wid] = pos_rk / block_rk;
    }
}

// ---------------------------------------------------------------------------
// Kernel 3: deterministic fixed-order reduction of the 3072 partials.
// out = 1 - (sum / (G*B))  where G*B = 3072.
// ---------------------------------------------------------------------------
__global__ void __launch_bounds__(256)
finalize_kernel(const float* __restrict__ partial, float* __restrict__ out)
{
    __shared__ float smem[256];
    float acc = 0.0f;
    for (int idx = threadIdx.x; idx < N_TRIPLE; idx += 256)
        acc += partial[idx];
    smem[threadIdx.x] = acc;
    __syncthreads();
    #pragma unroll
    for (int step = 128; step > 0; step >>= 1) {
        if ((int)threadIdx.x < step)
            smem[threadIdx.x] += smem[threadIdx.x + step];
        __syncthreads();
    }
    if (threadIdx.x == 0)
        out[0] = 1.0f - smem[0] / (float)N_TRIPLE;
}

// ---------------------------------------------------------------------------
extern "C" void kernel_launch(void* const* d_in, const int* in_sizes, int n_in,
                              void* d_out, int out_size, void* d_ws, size_t ws_size,
                              hipStream_t stream)
{
    (void)in_sizes; (void)n_in; (void)out_size; (void)ws_size;

    const float* preds = (const float*)d_in[0];
    float* sim     = (float*)d_ws;                 // 384*384 floats = 576 KB
    float* partial = sim + B_SZ * B_SZ;            // 3072 floats

    dim3 gemm_grid(B_SZ / 16, B_SZ / 16);          // 24 x 24 tiles
    simall_wmma_kernel<<<gemm_grid, 32, 0, stream>>>(preds, sim);

    // 3072 waves / 8 waves per block = 384 blocks
    rank_kernel<<<N_TRIPLE / 8, 256, 0, stream>>>(sim, partial);

    finalize_kernel<<<1, 256, 0, stream>>>(partial, (float*)d_out);
}